// SSIM_77120432767673
// MI455X (gfx1250) — compile-verified
//
#include <hip/hip_runtime.h>

// ---------------- geometry ----------------
#define H        1024
#define W        1024
#define NIMG     16
#define TILE     16
#define PATCH    26            // TILE + 2*5 halo
#define PROWS    28            // K padded to multiple of 4
#define PCOLS    32            // padded patch cols (bank friendly)
#define TTS      17            // transpose buffer stride (16+1)
#define WAVES    8             // waves per block, one 16x16 tile per wave
#define LDS_PER_WAVE (2*PROWS*PCOLS + PROWS*TTS)   // P1 + P2 + TT = 2268 floats
#define NBLOCKS  (NIMG * (H/TILE) * ((W/TILE)/WAVES))  // 16*64*8 = 8192

#define SSIM_C1  1.0e-4f
#define SSIM_C2  9.0e-4f

typedef float v2f __attribute__((ext_vector_type(2)));
typedef float v8f __attribute__((ext_vector_type(8)));

// normalized 11-tap gaussian, sigma=1.5 (precomputed, sums to 1.0)
__constant__ float GW[11] = {
    0.00102838f, 0.00759876f, 0.03600077f, 0.10936070f, 0.21300553f,
    0.26601172f,
    0.21300553f, 0.10936070f, 0.03600077f, 0.00759876f, 0.00102838f
};

__device__ __forceinline__ v8f wmma4(v2f a, v2f b, v8f c) {
    // D = A(16x4,f32) * B(4x16,f32) + C(16x16,f32)
    return __builtin_amdgcn_wmma_f32_16x16x4_f32(
        false, a, false, b, (short)0, c, false, false);
}

__device__ __forceinline__ float fieldval(const float* P1, const float* P2,
                                          int idx, int f) {
    float x = P1[idx];
    float y = P2[idx];
    switch (f) {
        case 0:  return x;
        case 1:  return y;
        case 2:  return x * x;
        case 3:  return y * y;
        default: return x * y;
    }
}

__global__ __launch_bounds__(WAVES * 32)
void ssim_tiles(const float* __restrict__ img1,
                const float* __restrict__ img2,
                float* __restrict__ blockSums) {
    __shared__ float lds[WAVES * LDS_PER_WAVE];
    __shared__ float wsum[WAVES];

    const int tid  = threadIdx.x;
    const int wave = tid >> 5;
    const int lane = tid & 31;

    // block -> (image, tile-row, tile-col-group); wave -> tile-col
    const int bid  = blockIdx.x;
    const int img  = bid >> 9;            // / (64*8)
    const int rem  = bid & 511;
    const int trow = rem >> 3;
    const int tgrp = rem & 7;
    const int tcol = tgrp * WAVES + wave;

    const int r0 = trow * TILE;
    const int c0 = tcol * TILE;
    const long base = (long)img * (H * W);

    float* slice = lds + wave * LDS_PER_WAVE;
    float* P1 = slice;                      // [PROWS][PCOLS]
    float* P2 = slice + PROWS * PCOLS;      // [PROWS][PCOLS]
    float* TT = slice + 2 * PROWS * PCOLS;  // [PROWS][TTS]

    // zero the whole per-wave slice once: gives K-padding + SAME zero-pad
    for (int i = lane; i < LDS_PER_WAVE; i += 32) slice[i] = 0.0f;

    // stage the 26x26 halo'd patches of both images
    for (int i = lane; i < PATCH * PATCH; i += 32) {
        int pr = i / PATCH, pc = i % PATCH;
        int gr = r0 - 5 + pr;
        int gc = c0 - 5 + pc;
        if (gr >= 0 && gr < H && gc >= 0 && gc < W) {
            long g = base + (long)gr * W + gc;
            P1[pr * PCOLS + pc] = img1[g];
            P2[pr * PCOLS + pc] = img2[g];
        }
    }

    // constant Toeplitz A fragments: Wv[m][k] = GW[k-m], 7 K-chunks of 4
    // A layout: lanes 0-15 hold M=lane, K = 4kc+{0,1}; lanes 16-31 K = 4kc+{2,3}
    const int halfHi = (lane >= 16) ? 2 : 0;
    const int m      = lane & 15;
    v2f afrag[7];
#pragma unroll
    for (int kc = 0; kc < 7; ++kc) {
        int k0 = 4 * kc + halfHi;
        int d0 = k0 - m, d1 = k0 + 1 - m;
        afrag[kc].x = (d0 >= 0 && d0 <= 10) ? GW[d0] : 0.0f;
        afrag[kc].y = (d1 >= 0 && d1 <= 10) ? GW[d1] : 0.0f;
    }

    const int n = lane & 15;
    v8f res[5];

#pragma unroll
    for (int f = 0; f < 5; ++f) {
        // ---- vertical pass: Tmp(16x26) = Wv(16x28) x Patch(28x26) ----
        v8f acc0 = {0.f,0.f,0.f,0.f,0.f,0.f,0.f,0.f};
        v8f acc1 = {0.f,0.f,0.f,0.f,0.f,0.f,0.f,0.f};
#pragma unroll
        for (int kc = 0; kc < 7; ++kc) {
            int kb = 4 * kc + halfHi;   // B rows (patch rows) for this lane half
            v2f b0, b1;
            b0.x = fieldval(P1, P2,  kb      * PCOLS + n,      f);
            b0.y = fieldval(P1, P2, (kb + 1) * PCOLS + n,      f);
            b1.x = fieldval(P1, P2,  kb      * PCOLS + 16 + n, f);
            b1.y = fieldval(P1, P2, (kb + 1) * PCOLS + 16 + n, f);
            acc0 = wmma4(afrag[kc], b0, acc0);
            acc1 = wmma4(afrag[kc], b1, acc1);
        }

        // ---- store Tmp transposed: TT[col][row] = Tmp[row][col] ----
#pragma unroll
        for (int t = 0; t < 2; ++t) {
            v8f acc = t ? acc1 : acc0;
#pragma unroll
            for (int i = 0; i < 8; ++i) {
                int row = i + ((lane < 16) ? 0 : 8);
                int col = 16 * t + n;
                if (col < PATCH) TT[col * TTS + row] = acc[i];
            }
        }

        // ---- horizontal pass (transposed): OutT = Wv(16x28) x TT(28x16) ----
        v8f acc = {0.f,0.f,0.f,0.f,0.f,0.f,0.f,0.f};
#pragma unroll
        for (int kc = 0; kc < 7; ++kc) {
            int kb = 4 * kc + halfHi;
            v2f b;
            b.x = TT[ kb      * TTS + n];
            b.y = TT[(kb + 1) * TTS + n];
            acc = wmma4(afrag[kc], b, acc);
        }
        res[f] = acc;   // transposed layout, identical for all 5 fields
    }

    // ---- elementwise SSIM + per-lane partial sum (8 pixels/lane) ----
    float sum = 0.0f;
#pragma unroll
    for (int i = 0; i < 8; ++i) {
        float a   = res[0][i];
        float b   = res[1][i];
        float e11 = res[2][i];
        float e22 = res[3][i];
        float e12 = res[4][i];
        float asq = a * a, bsq = b * b, ab = a * b;
        float s11 = e11 - asq;
        float s22 = e22 - bsq;
        float s12 = e12 - ab;
        float U = (2.0f * ab + SSIM_C1) * (2.0f * s12 + SSIM_C2);
        float V = (asq + bsq + SSIM_C1) * (s11 + s22 + SSIM_C2);
        float l = 1.0f - U / V;
        l = fminf(fmaxf(l, 0.0f), 1.0f);
        sum += 0.5f * l;
    }

    // wave32 reduction
#pragma unroll
    for (int off = 16; off > 0; off >>= 1)
        sum += __shfl_xor(sum, off, 32);

    if (lane == 0) wsum[wave] = sum;
    __syncthreads();
    if (tid == 0) {
        float t = 0.0f;
#pragma unroll
        for (int i = 0; i < WAVES; ++i) t += wsum[i];
        blockSums[blockIdx.x] = t;
    }
}

__global__ void ssim_reduce(const float* __restrict__ blockSums,
                            float* __restrict__ out) {
    __shared__ float red[256];
    int t = threadIdx.x;
    float s = 0.0f;
    for (int i = t; i < NBLOCKS; i += 256) s += blockSums[i];
    red[t] = s;
    __syncthreads();
    for (int off = 128; off > 0; off >>= 1) {
        if (t < off) red[t] += red[t + off];
        __syncthreads();
    }
    if (t == 0) out[0] = red[0] * (1.0f / ((float)NIMG * H * W));
}

extern "C" void kernel_launch(void* const* d_in, const int* in_sizes, int n_in,
                              void* d_out, int out_size, void* d_ws, size_t ws_size,
                              hipStream_t stream) {
    (void)in_sizes; (void)n_in; (void)out_size; (void)ws_size;
    const float* img1 = (const float*)d_in[0];
    const float* img2 = (const float*)d_in[1];
    float* out   = (float*)d_out;
    float* bsums = (float*)d_ws;   // NBLOCKS floats of scratch

    ssim_tiles<<<NBLOCKS, WAVES * 32, 0, stream>>>(img1, img2, bsums);
    ssim_reduce<<<1, 256, 0, stream>>>(bsums, out);
}